// GAT_5918464934679
// MI455X (gfx1250) — compile-verified
//
#include <hip/hip_runtime.h>
#include <math.h>

#define DIM 128
#define HEADS 4
#define CH 32
#define NEG_SLOPE 0.2f

typedef __attribute__((ext_vector_type(2))) float v2f;
typedef __attribute__((ext_vector_type(8))) float v8f;

// ---------------------------------------------------------------------------
// GEMM: out[N,128] = A[N,128] @ W[128,128] (+ bias broadcast over rows)
// One wave computes one 16x16 output tile with V_WMMA_F32_16X16X4_F32.
// Block = 256 threads = 8 waves -> 16 rows x 128 cols per block.
// ---------------------------------------------------------------------------
__global__ void gat_gemm_f32_wmma(const float* __restrict__ A,
                                  const float* __restrict__ W,
                                  const float* __restrict__ bias,
                                  float* __restrict__ out) {
  const int wave = threadIdx.x >> 5;
  const int lane = threadIdx.x & 31;
  const int row0 = blockIdx.x * 16;
  const int col0 = wave * 16;
  const int m  = lane & 15;   // matrix row within tile (A) / col (B,D)
  const int kh = lane >> 4;   // K half-select (lanes 16-31 hold K+2,K+3)

  const float* __restrict__ arow = A + (size_t)(row0 + m) * DIM;

  v8f acc = {};
  for (int k0 = 0; k0 < DIM; k0 += 4) {
    const int ka = k0 + kh * 2;
    v2f av, bv;
    // A (16x4 f32): element [m, k] lives in lane m + 16*(k/2), vgpr k%2
    av.x = arow[ka];
    av.y = arow[ka + 1];
    // B (4x16 f32): element [k, n] lives in lane n + 16*(k/2), vgpr k%2
    bv.x = W[(size_t)ka * DIM + col0 + m];
    bv.y = W[(size_t)(ka + 1) * DIM + col0 + m];
    acc = __builtin_amdgcn_wmma_f32_16x16x4_f32(
        false, av, false, bv, (short)0, acc, false, false);
  }

  // D (16x16 f32): vgpr r -> row r (lanes 0-15) / row r+8 (lanes 16-31), col = lane%16
  const int nc = col0 + m;
  const float b = bias ? bias[nc] : 0.0f;
  const int rbase = row0 + kh * 8;
#pragma unroll
  for (int r = 0; r < 8; ++r)
    out[(size_t)(rbase + r) * DIM + nc] = acc[r] + b;
}

// ---------------------------------------------------------------------------
// Per-(node, head) attention scores: asrc = <h[n,h,:], a_src[h,:]>, same for dst
// ---------------------------------------------------------------------------
__global__ void gat_attn_scores(const float* __restrict__ h,
                                const float* __restrict__ a_src,
                                const float* __restrict__ a_dst,
                                float* __restrict__ asrc,
                                float* __restrict__ adst, int N) {
  int t = blockIdx.x * blockDim.x + threadIdx.x;
  if (t >= N * HEADS) return;
  const int n = t >> 2, hh = t & 3;
  const float* __restrict__ hp = h + (size_t)n * DIM + hh * CH;
  const float* __restrict__ ws = a_src + hh * CH;
  const float* __restrict__ wd = a_dst + hh * CH;
  float s1 = 0.f, s2 = 0.f;
#pragma unroll
  for (int c = 0; c < CH; ++c) {
    const float v = hp[c];
    s1 += v * ws[c];
    s2 += v * wd[c];
  }
  asrc[t] = s1;
  adst[t] = s2;
}

__device__ __forceinline__ void edge_sd(const int* __restrict__ ei, int e,
                                        int NE, int& s, int& d) {
  if (e < NE) { s = ei[e]; d = ei[NE + e]; }
  else        { s = d = e - NE; }            // appended self-loops
}

__device__ __forceinline__ unsigned ord_encode(float f) {
  unsigned u = __float_as_uint(f);
  return (f >= 0.0f) ? (u | 0x80000000u) : ~u;  // monotonic float->uint map
}

// Pass 1: e = leaky_relu(asrc[src]+adst[dst]); atomicMax segment max (encoded)
__global__ void gat_edge_pass1(const int* __restrict__ ei, int NE, int E,
                               const float* __restrict__ asrc,
                               const float* __restrict__ adst,
                               float* __restrict__ ebuf,
                               unsigned* __restrict__ menc) {
  int t = blockIdx.x * blockDim.x + threadIdx.x;
  if (t >= E * HEADS) return;
  const int e = t >> 2, hh = t & 3;
  int s, d;
  edge_sd(ei, e, NE, s, d);
  float v = asrc[s * HEADS + hh] + adst[d * HEADS + hh];
  v = (v > 0.0f) ? v : NEG_SLOPE * v;
  ebuf[t] = v;
  atomicMax(&menc[d * HEADS + hh], ord_encode(v));
}

// Decode ordered-uint max back to float in place; 0 == "no edge" -> 0.0
__global__ void gat_decode_max(unsigned* __restrict__ menc, int n) {
  int t = blockIdx.x * blockDim.x + threadIdx.x;
  if (t >= n) return;
  const unsigned u = menc[t];
  float f;
  if (u == 0u)                 f = 0.0f;  // empty-segment guard (matches ref)
  else if (u & 0x80000000u)    f = __uint_as_float(u & 0x7FFFFFFFu);
  else                         f = __uint_as_float(~u);
  ((float*)menc)[t] = f;
}

// Pass 2: e <- exp(e - m[dst]); denom[dst] += e
__global__ void gat_edge_pass2(const int* __restrict__ ei, int NE, int E,
                               float* __restrict__ ebuf,
                               const float* __restrict__ m,
                               float* __restrict__ denom) {
  int t = blockIdx.x * blockDim.x + threadIdx.x;
  if (t >= E * HEADS) return;
  const int e = t >> 2, hh = t & 3;
  int s, d;
  edge_sd(ei, e, NE, s, d);
  const float ee = expf(ebuf[t] - m[d * HEADS + hh]);
  ebuf[t] = ee;
  atomicAdd(&denom[d * HEADS + hh], ee);
}

// Pass 3: agg[dst, c] += h[src, c] * (e / (denom[dst] + 1e-16)) per channel
__global__ void gat_edge_pass3(const int* __restrict__ ei, int NE, int E,
                               const float* __restrict__ ebuf,
                               const float* __restrict__ denom,
                               const float* __restrict__ h,
                               float* __restrict__ agg) {
  int t = blockIdx.x * blockDim.x + threadIdx.x;
  if (t >= E * DIM) return;
  const int e = t >> 7;          // edge
  const int c = t & (DIM - 1);   // channel 0..127
  const int hh = c >> 5;         // head = c / 32
  int s, d;
  edge_sd(ei, e, NE, s, d);
  const float alpha =
      ebuf[e * HEADS + hh] / (denom[d * HEADS + hh] + 1e-16f);
  atomicAdd(&agg[(size_t)d * DIM + c], h[(size_t)s * DIM + c] * alpha);
}

__global__ void gat_elu(const float* __restrict__ agg, float* __restrict__ out,
                        int n) {
  int t = blockIdx.x * blockDim.x + threadIdx.x;
  if (t >= n) return;
  const float v = agg[t];
  out[t] = (v > 0.0f) ? v : expm1f(v);
}

// ---------------------------------------------------------------------------
extern "C" void kernel_launch(void* const* d_in, const int* in_sizes, int n_in,
                              void* d_out, int out_size, void* d_ws,
                              size_t ws_size, hipStream_t stream) {
  const float* x  = (const float*)d_in[0];
  const int*   ei = (const int*)d_in[1];

  const int N  = in_sizes[0] / DIM;   // 50000
  const int NE = in_sizes[1] / 2;     // 600000
  const int E  = NE + N;              // edges + self-loops

  // Workspace carve-out
  float* ws = (float*)d_ws;
  size_t o = 0;
  float* h     = ws + o; o += (size_t)N * DIM;      // 25.6 MB
  float* agg   = ws + o; o += (size_t)N * DIM;      // 25.6 MB
  float* asrc  = ws + o; o += (size_t)N * HEADS;
  float* adst  = ws + o; o += (size_t)N * HEADS;
  float* mbuf  = ws + o; o += (size_t)N * HEADS;    // uint max then float
  float* denom = ws + o; o += (size_t)N * HEADS;
  float* ebuf  = ws + o; o += (size_t)E * HEADS;    // 10.4 MB

  float* y = (float*)d_out;

  const int TB = 256;
  const int gGemm = N / 16;                         // 50000 % 16 == 0
  const int gNH   = (N * HEADS + TB - 1) / TB;
  const int gEH   = (E * HEADS + TB - 1) / TB;
  const int gED   = (E * DIM + TB - 1) / TB;
  const int gND   = (N * DIM + TB - 1) / TB;

  const float* xin = x;
  for (int layer = 0; layer < 2; ++layer) {
    const float* W     = (const float*)d_in[2 + layer * 5];
    const float* a_src = (const float*)d_in[3 + layer * 5];
    const float* a_dst = (const float*)d_in[4 + layer * 5];
    const float* bias  = (const float*)d_in[5 + layer * 5];
    const float* Wskip = (const float*)d_in[6 + layer * 5];

    // h = x @ W ; agg = x @ Wskip + bias   (both via f32 WMMA)
    gat_gemm_f32_wmma<<<gGemm, TB, 0, stream>>>(xin, W, nullptr, h);
    gat_gemm_f32_wmma<<<gGemm, TB, 0, stream>>>(xin, Wskip, bias, agg);

    gat_attn_scores<<<gNH, TB, 0, stream>>>(h, a_src, a_dst, asrc, adst, N);

    hipMemsetAsync(mbuf, 0, (size_t)N * HEADS * sizeof(float), stream);
    hipMemsetAsync(denom, 0, (size_t)N * HEADS * sizeof(float), stream);

    gat_edge_pass1<<<gEH, TB, 0, stream>>>(ei, NE, E, asrc, adst, ebuf,
                                           (unsigned*)mbuf);
    gat_decode_max<<<gNH, TB, 0, stream>>>((unsigned*)mbuf, N * HEADS);
    gat_edge_pass2<<<gEH, TB, 0, stream>>>(ei, NE, E, ebuf, mbuf, denom);
    gat_edge_pass3<<<gED, TB, 0, stream>>>(ei, NE, E, ebuf, denom, h, agg);

    gat_elu<<<gND, TB, 0, stream>>>(agg, y, N * DIM);
    xin = y;  // layer 2 consumes layer 1 output in-place in d_out
  }
}